// CharRNNClassify_7868380086764
// MI455X (gfx1250) — compile-verified
//
#include <hip/hip_runtime.h>
#include <math.h>

// CDNA5 / gfx1250. wave32. GRU (2-layer, batch 1) + linear + log_softmax.
// Memory-bound fp32 GEMV implemented on V_WMMA_F32_16X16X4_F32.

typedef __attribute__((ext_vector_type(2))) float v2f;
typedef __attribute__((ext_vector_type(8))) float v8f;

#define H 4096
#define ROWS3H 12288  // 3*H

// ---------------------------------------------------------------------------
// GEMV: each wave-job computes a 16-row x 1024-K tile of W (row-major, row
// stride = S*1024) against vector x-slice, producing 16 partial sums stored
// at out[slice*12288 + row0 .. +16). Two matrices (ih, hh) per launch.
// ---------------------------------------------------------------------------
__global__ __launch_bounds__(256) void gemv16_wmma(
    const float* __restrict__ Wih, const float* __restrict__ xih,
    float* __restrict__ outih, int Sih,
    const float* __restrict__ Whh, const float* __restrict__ xhh,
    float* __restrict__ outhh, int Shh)
{
    __shared__ float xs[8 * 1024];              // 4KB x-slice per wave
    const int wid  = threadIdx.x >> 5;
    const int lane = threadIdx.x & 31;
    const int half = lane >> 4;                  // 0: lanes 0-15, 1: lanes 16-31
    const int row  = lane & 15;

    int job = blockIdx.x * 8 + wid;
    const float* W; const float* x; float* out; int S;
    const int nih = 768 * Sih;
    if (job < nih) { W = Wih; x = xih; out = outih; S = Sih; }
    else           { job -= nih; W = Whh; x = xhh; out = outhh; S = Shh; }
    const int rowblk = job / S;
    const int slice  = job % S;
    const int ld     = S * 1024;                 // row stride in floats
    const int row0   = rowblk * 16;

    // Stage this wave's 1024-float x-slice into its private LDS region.
    {
        const float4* xsrc = (const float4*)(x + slice * 1024);
        float4* xdst = (float4*)(xs + wid * 1024);
        #pragma unroll
        for (int i = 0; i < 8; ++i) xdst[lane + 32 * i] = xsrc[lane + 32 * i];
    }
    __syncthreads();

    const float4* wp = (const float4*)(W + (size_t)(row0 + row) * (size_t)ld
                                         + (size_t)slice * 1024);
    const float2* lx = (const float2*)(xs + wid * 1024);

    v8f c = {0.f, 0.f, 0.f, 0.f, 0.f, 0.f, 0.f, 0.f};

    // K loop: 8 K-values per iteration = 1 b128 weight load/lane + 2 WMMAs.
    // A layout (32-bit, 16x4): lane l<16 holds (k,k+1), lane l+16 holds
    // (k+2,k+3) of row l%16. Lane l loads W[row][k..k+3], lane l+16 loads
    // W[row][k+4..k+7]; one shfl_xor(16) pair redistributes the halves.
    #pragma unroll 4
    for (int k = 0; k < 1024; k += 8) {
        float4 wv = wp[(k >> 2) + half];
        float sa = half ? wv.x : wv.z;
        float sb = half ? wv.y : wv.w;
        float r0 = __shfl_xor(sa, 16, 32);
        float r1 = __shfl_xor(sb, 16, 32);

        v2f a0, a1;
        a0[0] = half ? r0   : wv.x;  a0[1] = half ? r1   : wv.y;  // K = k..k+3
        a1[0] = half ? wv.z : r0;    a1[1] = half ? wv.w : r1;    // K = k+4..k+7

        // B (4x16): broadcast x so every N column computes the same dot.
        // Per-lane B fragment = (x[k+2*half], x[k+1+2*half]).
        float2 xb0 = lx[(k >> 1) + half];
        float2 xb1 = lx[(k >> 1) + 2 + half];
        v2f b0, b1;
        b0[0] = xb0.x; b0[1] = xb0.y;
        b1[0] = xb1.x; b1[1] = xb1.y;

        c = __builtin_amdgcn_wmma_f32_16x16x4_f32(false, a0, false, b0,
                                                  (short)0, c, false, false);
        c = __builtin_amdgcn_wmma_f32_16x16x4_f32(false, a1, false, b1,
                                                  (short)0, c, false, false);
    }

    // D layout: lane 0 holds rows row0..row0+7 in c[0..7], lane 16 holds
    // rows row0+8..row0+15. Store partial sums (one writer per slot).
    if (row == 0) {
        float* dst = out + (size_t)slice * ROWS3H + row0 + half * 8;
        #pragma unroll
        for (int v = 0; v < 8; ++v) dst[v] = c[v];
    }
}

// ---------------------------------------------------------------------------
// Reduce K-slice partials and apply GRU gate math (PyTorch convention).
// ---------------------------------------------------------------------------
__global__ __launch_bounds__(256) void gru_combine(
    const float* __restrict__ gi, int Si,
    const float* __restrict__ gh, int Sh,
    const float* __restrict__ b_ih, const float* __restrict__ b_hh,
    const float* __restrict__ hprev, float* __restrict__ hout)
{
    const int m = blockIdx.x * blockDim.x + threadIdx.x;   // 0..H-1
    float gir = 0.f, giz = 0.f, gin = 0.f;
    for (int s = 0; s < Si; ++s) {
        const float* g = gi + (size_t)s * ROWS3H;
        gir += g[m]; giz += g[H + m]; gin += g[2 * H + m];
    }
    float ghr = 0.f, ghz = 0.f, ghn = 0.f;
    for (int s = 0; s < Sh; ++s) {
        const float* g = gh + (size_t)s * ROWS3H;
        ghr += g[m]; ghz += g[H + m]; ghn += g[2 * H + m];
    }
    float r = 1.f / (1.f + expf(-(gir + b_ih[m]         + ghr + b_hh[m])));
    float z = 1.f / (1.f + expf(-(giz + b_ih[H + m]     + ghz + b_hh[H + m])));
    float n = tanhf(gin + b_ih[2 * H + m] + r * (ghn + b_hh[2 * H + m]));
    hout[m] = (1.f - z) * n + z * hprev[m];
}

// ---------------------------------------------------------------------------
// 9x4096 GEMV + bias + log_softmax (single workgroup).
// ---------------------------------------------------------------------------
__global__ __launch_bounds__(256) void linear_logsoftmax(
    const float* __restrict__ lin_w, const float* __restrict__ lin_b,
    const float* __restrict__ h1, float* __restrict__ logp)
{
    __shared__ float red[9][256];
    float acc[9];
    #pragma unroll
    for (int o = 0; o < 9; ++o) acc[o] = 0.f;
    for (int k = threadIdx.x; k < H; k += 256) {
        float hv = h1[k];
        #pragma unroll
        for (int o = 0; o < 9; ++o) acc[o] += lin_w[o * H + k] * hv;
    }
    #pragma unroll
    for (int o = 0; o < 9; ++o) red[o][threadIdx.x] = acc[o];
    __syncthreads();
    for (int s = 128; s > 0; s >>= 1) {
        if (threadIdx.x < (unsigned)s) {
            #pragma unroll
            for (int o = 0; o < 9; ++o)
                red[o][threadIdx.x] += red[o][threadIdx.x + s];
        }
        __syncthreads();
    }
    if (threadIdx.x == 0) {
        float lg[9]; float mx = -3.0e38f;
        #pragma unroll
        for (int o = 0; o < 9; ++o) { lg[o] = red[o][0] + lin_b[o]; mx = fmaxf(mx, lg[o]); }
        float s = 0.f;
        #pragma unroll
        for (int o = 0; o < 9; ++o) s += expf(lg[o] - mx);
        float ls = logf(s);
        #pragma unroll
        for (int o = 0; o < 9; ++o) logp[o] = lg[o] - mx - ls;
    }
}

extern "C" void kernel_launch(void* const* d_in, const int* in_sizes, int n_in,
                              void* d_out, int out_size, void* d_ws, size_t ws_size,
                              hipStream_t stream) {
    const float* input  = (const float*)d_in[0];   // [1,1024]
    const float* hidden = (const float*)d_in[1];   // [2,1,4096]
    const float* w_ih0  = (const float*)d_in[2];   // [12288,1024]
    const float* w_hh0  = (const float*)d_in[3];   // [12288,4096]
    const float* b_ih0  = (const float*)d_in[4];
    const float* b_hh0  = (const float*)d_in[5];
    const float* w_ih1  = (const float*)d_in[6];   // [12288,4096]
    const float* w_hh1  = (const float*)d_in[7];   // [12288,4096]
    const float* b_ih1  = (const float*)d_in[8];
    const float* b_hh1  = (const float*)d_in[9];
    const float* lin_w  = (const float*)d_in[10];  // [9,4096]
    const float* lin_b  = (const float*)d_in[11];

    float* out  = (float*)d_out;
    float* logp = out;                 // [9]
    float* h0   = out + 9;             // [4096]
    float* h1   = out + 9 + H;         // [4096]

    // Workspace: gi partials (up to 4 slices) then gh partials (4 slices).
    float* gi = (float*)d_ws;                       // 4*12288 floats
    float* gh = gi + 4 * ROWS3H;                    // 4*12288 floats (total 384KB)

    // Layer 0: w_ih0 K=1024 (1 slice), w_hh0 K=4096 (4 slices).
    // jobs = 768*1 + 768*4 = 3840 waves = 480 blocks of 8 waves.
    gemv16_wmma<<<480, 256, 0, stream>>>(w_ih0, input, gi, 1,
                                         w_hh0, hidden, gh, 4);
    gru_combine<<<H / 256, 256, 0, stream>>>(gi, 1, gh, 4, b_ih0, b_hh0,
                                             hidden, h0);

    // Layer 1: both K=4096 (4 slices each). jobs = 6144 waves = 768 blocks.
    gemv16_wmma<<<768, 256, 0, stream>>>(w_ih1, h0, gi, 4,
                                         w_hh1, hidden + H, gh, 4);
    gru_combine<<<H / 256, 256, 0, stream>>>(gi, 4, gh, 4, b_ih1, b_hh1,
                                             hidden + H, h1);

    linear_logsoftmax<<<1, 256, 0, stream>>>(lin_w, lin_b, h1, logp);
}